// GCNWithEmbeddings_44976897524562
// MI455X (gfx1250) — compile-verified
//
#include <hip/hip_runtime.h>

// ---------------------------------------------------------------------------
// GCN 2-layer forward for MI455X (gfx1250, wave32, WMMA).
//   h1 = x @ W1            (WMMA f32 16x16x4, W1 pre-swizzled in LDS)
//   a1 = GCN-agg(h1) + b1  (self-loop init + fp32 atomic edge scatter)
//   h2 = relu(a1) @ W2     (WMMA, relu fused into A load)
//   out = GCN-agg(h2) + b2
// ---------------------------------------------------------------------------

typedef __attribute__((ext_vector_type(2))) float v2f;
typedef __attribute__((ext_vector_type(8))) float v8f;

#define IN_DIM  128
#define HID_DIM 64
#define OUT_DIM 32

// ---- degree / normalization ------------------------------------------------
__global__ void deg_init_kernel(float* __restrict__ deg, int n) {
  int i = blockIdx.x * blockDim.x + threadIdx.x;
  if (i < n) deg[i] = 1.0f;  // self-loop
}

__global__ void deg_accum_kernel(const long long* __restrict__ dst,
                                 float* __restrict__ deg, int e) {
  int i = blockIdx.x * blockDim.x + threadIdx.x;
  if (i < e) atomicAdd(&deg[(int)dst[i]], 1.0f);
}

__global__ void dinv_kernel(float* __restrict__ deg, int n) {
  int i = blockIdx.x * blockDim.x + threadIdx.x;
  if (i < n) deg[i] = __frsqrt_rn(deg[i]);  // deg >= 1 always
}

// ---- WMMA GEMM: D[nrows x NC] = (RELU?relu:id)(A[nrows x K]) * W[K x NC] ---
// Block = 256 threads = 8 waves; each wave owns a 16-row stripe and all NC
// columns (NC/16 accumulators), so the per-lane A fragment is reused NC/16x.
//
// fp32 16x16x4 A layout (ISA 7.12.2): lanes 0-15 hold {K=0,K=1} of row M=lane,
// lanes 16-31 hold {K=2,K=3} of row M=lane-16  -> one global b64 load per lane.
// B (4x16) mirrors it: lane supplies rows {k+2*half, k+2*half+1} of column
// l16. W is staged into LDS *pre-swizzled into fragment order* so each lane's
// B pair is one contiguous 8-byte element -> single ds_load_b64 per fragment,
// no VGPR repacking between WMMAs.
template <int K, int NC, bool RELU>
__global__ __launch_bounds__(256) void gemm_wmma_kernel(
    const float* __restrict__ A, const float* __restrict__ W,
    float* __restrict__ D, int nrows) {
  constexpr int NCT   = NC / 16;       // column tiles
  constexpr int KS    = K / 4;         // k-steps
  constexpr int PAIRS = KS * NCT * 32; // one v2f per (kstep, ctile, lane)
  __shared__ v2f sW[PAIRS];

  const int tid = threadIdx.x;
  // Stage W in fragment order: sW[(ks*NCT + c)*32 + lane] = {W[k][n], W[k+1][n]}
  // with k = 4*ks + 2*(lane>>4), n = c*16 + (lane&15).
  for (int j = tid; j < PAIRS; j += 256) {
    const int ks  = j / (NCT * 32);
    const int rem = j % (NCT * 32);
    const int c   = rem >> 5;
    const int L   = rem & 31;
    const int k   = 4 * ks + 2 * (L >> 4);
    const int n   = c * 16 + (L & 15);
    v2f p;
    p.x = W[k * NC + n];
    p.y = W[(k + 1) * NC + n];
    sW[j] = p;
  }
  __syncthreads();

  const int wave = tid >> 5;
  const int lane = tid & 31;
  const int half = lane >> 4;   // 0: lanes 0-15, 1: lanes 16-31
  const int l16  = lane & 15;
  const int m0   = (blockIdx.x * 8 + wave) * 16;
  if (m0 >= nrows) return;      // wave-uniform exit (EXEC stays all-1 for WMMA)

  int mrow = m0 + l16;
  if (mrow >= nrows) mrow = nrows - 1;  // clamped tail loads; stores are guarded
  const float* __restrict__ arow = A + (size_t)mrow * K + 2 * half;

  v8f acc[NCT] = {};

  for (int ks = 0; ks < KS; ++ks) {
    v2f a;
    a.x = arow[4 * ks];
    a.y = arow[4 * ks + 1];
    if (RELU) {
      a.x = fmaxf(a.x, 0.0f);
      a.y = fmaxf(a.y, 0.0f);
    }
#pragma unroll
    for (int c = 0; c < NCT; ++c) {
      v2f b = sW[(ks * NCT + c) * 32 + lane];  // ds_load_b64, conflict-free
      acc[c] = __builtin_amdgcn_wmma_f32_16x16x4_f32(
          /*neg_a=*/false, a, /*neg_b=*/false, b,
          /*c_mod=*/(short)0, acc[c], /*reuse_a=*/false, /*reuse_b=*/false);
    }
  }

  // C/D layout: VGPR r = row (m0 + r + 8*half), column l16.
#pragma unroll
  for (int r = 0; r < 8; ++r) {
    int row = m0 + r + 8 * half;
    if (row < nrows) {
#pragma unroll
      for (int c = 0; c < NCT; ++c)
        D[(size_t)row * NC + c * 16 + l16] = acc[c][r];
    }
  }
}

// ---- aggregation: self-loop term + bias ------------------------------------
template <int F>
__global__ void agg_init_kernel(const float* __restrict__ h,
                                const float* __restrict__ dinv,
                                const float* __restrict__ bias,
                                float* __restrict__ out, int n) {
  unsigned t = blockIdx.x * blockDim.x + threadIdx.x;
  if (t < (unsigned)n * F) {
    unsigned i = t / F;
    unsigned f = t % F;  // F is a power of two -> shifts/masks
    float di = dinv[i];
    out[t] = h[t] * di * di + bias[f];
  }
}

// ---- edge scatter: out[d] += h[s] * dinv[s]*dinv[d] ------------------------
// F consecutive threads share one edge: coalesced gather of h[s,:], converged
// index loads, fp32 atomics into L2-resident destination rows.
template <int F>
__global__ void scatter_kernel(const float* __restrict__ h,
                               float* __restrict__ out,
                               const long long* __restrict__ src,
                               const long long* __restrict__ dst,
                               const float* __restrict__ dinv, unsigned e) {
  unsigned t = blockIdx.x * blockDim.x + threadIdx.x;
  if (t >= e * F) return;
  unsigned ei = t / F;
  unsigned f  = t % F;
  int s = (int)src[ei];
  int d = (int)dst[ei];
  float coef = dinv[s] * dinv[d];
  atomicAdd(&out[(unsigned)d * F + f], h[(unsigned)s * F + f] * coef);
}

// ---------------------------------------------------------------------------
extern "C" void kernel_launch(void* const* d_in, const int* in_sizes, int n_in,
                              void* d_out, int out_size, void* d_ws,
                              size_t ws_size, hipStream_t stream) {
  const float*     x   = (const float*)d_in[0];
  const long long* ei  = (const long long*)d_in[1];  // int64 [2, E]
  const float*     W1  = (const float*)d_in[2];
  const float*     b1  = (const float*)d_in[3];
  const float*     W2  = (const float*)d_in[4];
  const float*     b2  = (const float*)d_in[5];
  float*           out = (float*)d_out;

  const int N = in_sizes[0] / IN_DIM;
  const int E = in_sizes[1] / 2;
  const long long* src = ei;
  const long long* dst = ei + E;

  // Workspace layout (floats): dinv[N] | h1[N*64] | a1[N*64] | h2[N*32]
  float* dinv = (float*)d_ws;
  float* h1   = dinv + N;
  float* a1   = h1 + (size_t)N * HID_DIM;
  float* h2   = a1 + (size_t)N * HID_DIM;

  const int T = 256;
  // 1) symmetric normalization coefficients
  deg_init_kernel<<<(N + T - 1) / T, T, 0, stream>>>(dinv, N);
  deg_accum_kernel<<<(E + T - 1) / T, T, 0, stream>>>(dst, dinv, E);
  dinv_kernel<<<(N + T - 1) / T, T, 0, stream>>>(dinv, N);

  // 2) layer 1: GEMM -> init(self-loop+bias) -> edge scatter
  gemm_wmma_kernel<IN_DIM, HID_DIM, false>
      <<<(N + 127) / 128, T, 0, stream>>>(x, W1, h1, N);
  {
    size_t tot = (size_t)N * HID_DIM;
    agg_init_kernel<HID_DIM>
        <<<(unsigned)((tot + T - 1) / T), T, 0, stream>>>(h1, dinv, b1, a1, N);
    size_t work = (size_t)E * HID_DIM;
    scatter_kernel<HID_DIM>
        <<<(unsigned)((work + T - 1) / T), T, 0, stream>>>(h1, a1, src, dst,
                                                           dinv, (unsigned)E);
  }

  // 3) layer 2: relu fused into GEMM A-load -> init -> edge scatter into d_out
  gemm_wmma_kernel<HID_DIM, OUT_DIM, true>
      <<<(N + 127) / 128, T, 0, stream>>>(a1, W2, h2, N);
  {
    size_t tot = (size_t)N * OUT_DIM;
    agg_init_kernel<OUT_DIM>
        <<<(unsigned)((tot + T - 1) / T), T, 0, stream>>>(h2, dinv, b2, out, N);
    size_t work = (size_t)E * OUT_DIM;
    scatter_kernel<OUT_DIM>
        <<<(unsigned)((work + T - 1) / T), T, 0, stream>>>(h2, out, src, dst,
                                                           dinv, (unsigned)E);
  }
}